// generator_gnn_38302518345767
// MI455X (gfx1250) — compile-verified
//
#include <hip/hip_runtime.h>
#include <cstddef>

typedef float v2f __attribute__((ext_vector_type(2)));
typedef float v8f __attribute__((ext_vector_type(8)));

#define NB 10000
#define BB 8
#define EE 160000
#define EER 150000        // random edges; edges [EER, EE) are self-loops (row=col=n, w=1)
#define MM (BB * NB)      // 80000 rows for the dense stage
#define ZK 134
#define WSTRIDE 160       // LDS row stride per K-pair: 160 % 64 == 32 -> bank-half split

// ---------------------------------------------------------------------------
// h0[n*8+b] = b_map[n] + sum_k z[b][k] * W_map[k*N+n]
// z = concat(x, emb0[cc0], emb1[cc1]) built in LDS.
// ---------------------------------------------------------------------------
__global__ void k_h0(const float* __restrict__ x, const int* __restrict__ cc,
                     const float* __restrict__ emb0, const float* __restrict__ emb1,
                     const float* __restrict__ W_map, const float* __restrict__ b_map,
                     float* __restrict__ h0) {
  __shared__ float z[BB][ZK];
  int tid = threadIdx.x;
  for (int i = tid; i < BB * ZK; i += 256) {
    int b = i / ZK, k = i % ZK;
    float v;
    if (k < 128)      v = x[b * 128 + k];
    else if (k < 130) v = emb0[cc[b * 2 + 0] * 2 + (k - 128)];
    else              v = emb1[cc[b * 2 + 1] * 4 + (k - 130)];
    z[b][k] = v;
  }
  __syncthreads();
  int n = blockIdx.x * 256 + tid;
  if (n >= NB) return;
  float bm = b_map[n];
  float acc[BB];
#pragma unroll
  for (int b = 0; b < BB; ++b) acc[b] = bm;
  for (int k = 0; k < ZK; ++k) {
    float wm = W_map[(size_t)k * NB + n];   // coalesced across n
#pragma unroll
    for (int b = 0; b < BB; ++b) acc[b] += z[b][k] * wm;
  }
#pragma unroll
  for (int b = 0; b < BB; ++b) h0[n * BB + b] = acc[b];
}

// ---------------------------------------------------------------------------
// degree + symmetric norm (over ALL edges, incl. self-loops)
// ---------------------------------------------------------------------------
__global__ void k_deg(const int* __restrict__ col, const float* __restrict__ w,
                      float* __restrict__ deg) {
  int t = blockIdx.x * 256 + threadIdx.x;
  if (t >= EE) return;
  atomicAdd(&deg[col[t]], w[t]);
}

__global__ void k_norm(const int* __restrict__ row, const int* __restrict__ col,
                       const float* __restrict__ w, const float* __restrict__ deg,
                       float* __restrict__ norm) {
  int t = blockIdx.x * 256 + threadIdx.x;
  if (t >= EE) return;
  float dr = deg[row[t]], dc = deg[col[t]];
  float ir = dr > 0.f ? rsqrtf(dr) : 0.f;
  float ic = dc > 0.f ? rsqrtf(dc) : 0.f;
  norm[t] = ir * w[t] * ic;
}

// ---------------------------------------------------------------------------
// Layer-0 propagation (1 channel). Self-loop folded into init, atomics only
// for the EER random edges.
// ---------------------------------------------------------------------------
__global__ void k_s0init(const float* __restrict__ h0, const float* __restrict__ norm,
                         float* __restrict__ s0) {
  int t = blockIdx.x * 256 + threadIdx.x;
  if (t >= MM) return;
  int n = t >> 3;                     // layout [n*8+b]
  s0[t] = h0[t] * norm[EER + n];
}

__global__ void k_scatter1(const int* __restrict__ row, const int* __restrict__ col,
                           const float* __restrict__ norm, const float* __restrict__ src,
                           float* __restrict__ dst) {
  int t = blockIdx.x * 256 + threadIdx.x;
  if (t >= EER * BB) return;
  int b = t & 7, e = t >> 3;
  float v = src[row[e] * BB + b] * norm[e];
  atomicAdd(&dst[col[e] * BB + b], v);
}

// ---------------------------------------------------------------------------
// h1[m*128+c] = relu(s0 * W0[c] + b0[c]),  m = b*N + n
// ---------------------------------------------------------------------------
__global__ void k_expand(const float* __restrict__ s0, const float* __restrict__ W0,
                         const float* __restrict__ b0, float* __restrict__ h1) {
  int t = blockIdx.x * 256 + threadIdx.x;  // exactly MM*128 threads
  int c = t & 127, m = t >> 7;
  int b = m / NB, n = m - b * NB;
  float v = s0[n * BB + b] * W0[c] + b0[c];
  h1[t] = v > 0.f ? v : 0.f;
}

// ---------------------------------------------------------------------------
// Layer-1 GEMM: h2pre(80000x64) = h1(80000x128) @ W1(128x64), f32 WMMA 16x16x4.
// One wave per 16-row M-tile, covering all 4 N-tiles (64 cols).
// W1 staged interleaved in LDS: sW1[kpair*WSTRIDE + 2*n + (k&1)], so each B
// fragment (K=kb,kb+1 at column n) is one aligned ds_load_b64.
// A frag (16x4 f32): lane m=l&15; lanes16-31 hold K+2/K+3 in vgpr0/1.
// C/D (16x16 f32): vgpr j -> M = (l>>4)*8 + j, N = l&15.
// ---------------------------------------------------------------------------
__global__ void k_gemm1(const float* __restrict__ h1, const float* __restrict__ W1,
                        float* __restrict__ h2pre) {
  __shared__ float sW1[64 * WSTRIDE];       // 40 KB
  int tid = threadIdx.x;
  for (int i = tid; i < 128 * 64; i += 256) {
    int k = i >> 6, n = i & 63;
    sW1[(k >> 1) * WSTRIDE + 2 * n + (k & 1)] = W1[i];
  }
  __syncthreads();

  int wave = tid >> 5, lane = tid & 31;
  int tileM = blockIdx.x * 8 + wave;        // grid=625 -> exactly 5000 tiles
  int m0 = tileM * 16;
  int mrow = m0 + (lane & 15);
  int khalf = (lane >> 4) * 2;              // 0 or 2
  int ncol = lane & 15;
  const float* Arow = h1 + (size_t)mrow * 128;

  v8f acc0 = {}, acc1 = {}, acc2 = {}, acc3 = {};
  for (int k = 0; k < 128; k += 4) {
    int kb = k + khalf;                     // even
    v2f a;
    a.x = Arow[kb];
    a.y = Arow[kb + 1];
    const float* brow = &sW1[(kb >> 1) * WSTRIDE + 2 * ncol];
    v2f b0v = *(const v2f*)(brow + 0);
    v2f b1v = *(const v2f*)(brow + 32);
    v2f b2v = *(const v2f*)(brow + 64);
    v2f b3v = *(const v2f*)(brow + 96);
    acc0 = __builtin_amdgcn_wmma_f32_16x16x4_f32(false, a, false, b0v, (short)0, acc0, false, false);
    acc1 = __builtin_amdgcn_wmma_f32_16x16x4_f32(false, a, false, b1v, (short)0, acc1, false, false);
    acc2 = __builtin_amdgcn_wmma_f32_16x16x4_f32(false, a, false, b2v, (short)0, acc2, false, false);
    acc3 = __builtin_amdgcn_wmma_f32_16x16x4_f32(false, a, false, b3v, (short)0, acc3, false, false);
  }

  int mb = (lane >> 4) * 8;
  float* out = h2pre + (size_t)(m0 + mb) * 64 + ncol;
#pragma unroll
  for (int j = 0; j < 8; ++j) {
    out[(size_t)j * 64 + 0]  = acc0[j];
    out[(size_t)j * 64 + 16] = acc1[j];
    out[(size_t)j * 64 + 32] = acc2[j];
    out[(size_t)j * 64 + 48] = acc3[j];
  }
}

// ---------------------------------------------------------------------------
// Layer-1 propagation (64 channels). Self-loop contribution is the init
// (replaces the 20.5MB memset); atomics only for the EER random edges.
// ---------------------------------------------------------------------------
__global__ void k_h2init(const float* __restrict__ h2pre, const float* __restrict__ norm,
                         float* __restrict__ h2) {
  int t = blockIdx.x * 256 + threadIdx.x;   // exactly MM*64 threads
  int m = t >> 6;                           // b*NB + n
  int n = m % NB;
  h2[t] = h2pre[t] * norm[EER + n];
}

__global__ void k_scatter2(const int* __restrict__ row, const int* __restrict__ col,
                           const float* __restrict__ norm, const float* __restrict__ h2pre,
                           float* __restrict__ h2) {
  int t = blockIdx.x * 256 + threadIdx.x;   // exactly EER*512 threads
  int c = t & 63;
  int b = (t >> 6) & 7;
  int e = t >> 9;
  float v = h2pre[((size_t)(b * NB + row[e])) * 64 + c] * norm[e];
  atomicAdd(&h2[((size_t)(b * NB + col[e])) * 64 + c], v);
}

// ---------------------------------------------------------------------------
// t2[n*8+b] = sum_c relu(h2[m,c]+b1[c]) * W2[c]   (one wave per row, shuffle-reduce)
// ---------------------------------------------------------------------------
__global__ void k_t2(const float* __restrict__ h2, const float* __restrict__ b1,
                     const float* __restrict__ W2, float* __restrict__ t2) {
  int wave = threadIdx.x >> 5, lane = threadIdx.x & 31;
  int m = blockIdx.x * 8 + wave;            // grid=10000 -> exactly 80000 rows
  const float* hrow = h2 + (size_t)m * 64;
  float v0 = hrow[lane] + b1[lane];           v0 = v0 > 0.f ? v0 : 0.f;
  float v1 = hrow[lane + 32] + b1[lane + 32]; v1 = v1 > 0.f ? v1 : 0.f;
  float v = v0 * W2[lane] + v1 * W2[lane + 32];
#pragma unroll
  for (int off = 16; off > 0; off >>= 1) v += __shfl_xor(v, off, 32);
  if (lane == 0) {
    int b = m / NB, n = m - b * NB;
    t2[n * BB + b] = v;
  }
}

// ---------------------------------------------------------------------------
// Layer-2: out init = b2 + self-loop term, then 1-channel scatter (random edges)
// out layout is the reference (B,N) b-major.
// ---------------------------------------------------------------------------
__global__ void k_fill(const float* __restrict__ b2, const float* __restrict__ t2,
                       const float* __restrict__ norm, float* __restrict__ out) {
  int t = blockIdx.x * 256 + threadIdx.x;
  if (t >= MM) return;
  int b = t / NB, n = t - b * NB;
  out[t] = b2[0] + t2[n * BB + b] * norm[EER + n];
}

__global__ void k_scatter3(const int* __restrict__ row, const int* __restrict__ col,
                           const float* __restrict__ norm, const float* __restrict__ t2,
                           float* __restrict__ out) {
  int t = blockIdx.x * 256 + threadIdx.x;
  if (t >= EER * BB) return;
  int b = t & 7, e = t >> 3;
  float v = t2[row[e] * BB + b] * norm[e];
  atomicAdd(&out[b * NB + col[e]], v);
}

// ---------------------------------------------------------------------------
extern "C" void kernel_launch(void* const* d_in, const int* in_sizes, int n_in,
                              void* d_out, int out_size, void* d_ws, size_t ws_size,
                              hipStream_t stream) {
  const float* x     = (const float*)d_in[0];
  const int*   cc    = (const int*)  d_in[1];
  const int*   erow  = (const int*)  d_in[2];
  const int*   ecol  = (const int*)  d_in[3];
  const float* ew    = (const float*)d_in[4];
  const float* emb0  = (const float*)d_in[5];
  const float* emb1  = (const float*)d_in[6];
  const float* W_map = (const float*)d_in[7];
  const float* b_map = (const float*)d_in[8];
  const float* W0    = (const float*)d_in[9];
  const float* b0    = (const float*)d_in[10];
  const float* W1    = (const float*)d_in[11];
  const float* b1    = (const float*)d_in[12];
  const float* W2    = (const float*)d_in[13];
  const float* b2    = (const float*)d_in[14];
  float* out = (float*)d_out;

  // workspace layout (float offsets)
  float* ws = (float*)d_ws;
  float* deg   = ws;                 //     10240
  float* norm  = ws + 10240;         //   160000
  float* h0    = ws + 170240;        //    80000
  float* s0    = ws + 250240;        //    80000
  float* t2    = ws + 330240;        //    80000
  float* h2pre = ws + 410240;        //  5120000
  float* h1    = ws + 5530240;       // 10240000  (h2 aliases h1 after GEMM)
  float* h2    = h1;

  // degree + norm
  hipMemsetAsync(deg, 0, NB * sizeof(float), stream);
  k_deg <<<(EE + 255) / 256, 256, 0, stream>>>(ecol, ew, deg);
  k_norm<<<(EE + 255) / 256, 256, 0, stream>>>(erow, ecol, ew, deg, norm);

  // dense map -> h0 (1 channel per node)
  k_h0<<<(NB + 255) / 256, 256, 0, stream>>>(x, cc, emb0, emb1, W_map, b_map, h0);

  // layer 0: 1-channel propagation (self-loop folded into init), expand to 128 + relu
  k_s0init  <<<(MM + 255) / 256, 256, 0, stream>>>(h0, norm, s0);
  k_scatter1<<<(EER * BB + 255) / 256, 256, 0, stream>>>(erow, ecol, norm, h0, s0);
  k_expand  <<<(MM * 128) / 256, 256, 0, stream>>>(s0, W0, b0, h1);

  // layer 1: WMMA GEMM 128->64, then 64-channel propagation
  k_gemm1   <<<625, 256, 0, stream>>>(h1, W1, h2pre);
  k_h2init  <<<(MM * 64) / 256, 256, 0, stream>>>(h2pre, norm, h2);
  k_scatter2<<<(EER * 8 * 64) / 256, 256, 0, stream>>>(erow, ecol, norm, h2pre, h2);

  // layer 2: bias+relu fused into 64->1 contraction, then 1-channel propagation
  k_t2      <<<MM / 8, 256, 0, stream>>>(h2, b1, W2, t2);
  k_fill    <<<(MM + 255) / 256, 256, 0, stream>>>(b2, t2, norm, out);
  k_scatter3<<<(EER * BB + 255) / 256, 256, 0, stream>>>(erow, ecol, norm, t2, out);
}